// Net_26912265076915
// MI455X (gfx1250) — compile-verified
//
#include <hip/hip_runtime.h>

// ---------------------------------------------------------------------------
// Submanifold sparse 3D conv backbone on MI455X (gfx1250).
// GEMMs: V_WMMA_F32_16X16X4_F32 (exact fp32 on the matrix pipe), templated on
// (CIN,COUT) so the K-loop fully unrolls into constant-offset loads + WMMA.
// Each wave owns a 16x32 output tile (2 WMMA accumulators sharing one A op).
// Gather of the next kernel-offset's 16xCIN A-tile is software-pipelined
// through registers into a double-buffered LDS tile (one barrier per offset).
// All structure (neighbor pairs, pooling) built on-device; launches use
// host-side upper bounds + device-side counts -> graph-capture safe.
// ---------------------------------------------------------------------------

typedef __attribute__((ext_vector_type(2))) float v2f;
typedef __attribute__((ext_vector_type(8))) float v8f;

__global__ void k_init_n(int* dn, int n0) {
  if (threadIdx.x == 0) dn[0] = n0;
}

// grid[lin(coord)] = row   (grid pre-memset to -1)
__global__ void k_scatter_grid(const int* __restrict__ coords,
                               int* __restrict__ grid,
                               const int* __restrict__ pn, int S) {
  int i = blockIdx.x * 256 + threadIdx.x;
  if (i >= *pn) return;
  int z = coords[i * 4 + 1], y = coords[i * 4 + 2], x = coords[i * 4 + 3];
  grid[(z * S + y) * S + x] = i;
}

// pairs[k][row] = neighbor row for kernel offset k, or -1
__global__ void k_build_pairs(const int* __restrict__ coords,
                              const int* __restrict__ grid,
                              int* __restrict__ pairs,
                              const int* __restrict__ pn, int n_ub, int S) {
  int i = blockIdx.x * 256 + threadIdx.x;
  int k = blockIdx.y;
  if (i >= *pn) return;
  int dz = k / 9 - 1, dy = (k / 3) % 3 - 1, dx = k % 3 - 1;
  int z = coords[i * 4 + 1] + dz;
  int y = coords[i * 4 + 2] + dy;
  int x = coords[i * 4 + 3] + dx;
  int r = -1;
  if (z >= 0 && z < S && y >= 0 && y < S && x >= 0 && x < S)
    r = grid[(z * S + y) * S + x];
  pairs[k * n_ub + i] = r;
}

// ------------------------------ conv ---------------------------------------

template <int CIN, int COUT>
__global__ __launch_bounds__(128)
void k_conv_wmma(const float* __restrict__ fin, const float* __restrict__ W,
                 float* __restrict__ fout, const int* __restrict__ pairs,
                 const int* __restrict__ pn, int n_ub) {
  constexpr int CINP = (CIN % 2) ? (CIN + 5) : (CIN + 4);  // even, +pad vs banks
  int n = *pn;
  int rowbase = blockIdx.x * 16;
  if (rowbase >= n) return;                    // uniform over block

  extern __shared__ float smem[];
  float* const bufA = smem;
  float* const bufB = smem + 16 * CINP;

  const int tid  = threadIdx.x;
  const int wv   = tid >> 5;
  const int lane = tid & 31;
  const int hlf  = lane >> 4;                  // ISA lane-half (K/M split)
  const int m    = lane & 15;
  const int cb   = blockIdx.y * 128 + wv * 32; // 32 output cols per wave
  const bool active = (cb < COUT);             // wave-uniform

  v8f acc0 = {0.f, 0.f, 0.f, 0.f, 0.f, 0.f, 0.f, 0.f};
  v8f acc1 = {0.f, 0.f, 0.f, 0.f, 0.f, 0.f, 0.f, 0.f};

  // ---- register-staged gather of one 16xCIN A tile -------------------------
  constexpr bool VEC2 = (CIN % 16 == 0);
  constexpr int NS = VEC2 ? (CIN / 16) : ((16 * CIN + 127) / 128);
  v2f   rg2[VEC2 ? NS : 1];
  float rg1[VEC2 ? 1 : NS];

  auto ldStage = [&](int k) {
    const int* __restrict__ pk = pairs + k * n_ub + rowbase;
    if constexpr (VEC2) {
#pragma unroll
      for (int i = 0; i < NS; ++i) {
        int e2 = tid + i * 128;
        int rm = e2 / (CIN / 2);
        int c2 = e2 - rm * (CIN / 2);
        int row = rowbase + rm;
        int id = (row < n) ? pk[rm] : -1;
        v2f v = {0.f, 0.f};
        if (id >= 0) v = *(const v2f*)(fin + (size_t)id * CIN + c2 * 2);
        rg2[i] = v;
      }
    } else {
#pragma unroll
      for (int i = 0; i < NS; ++i) {
        int e = tid + i * 128;
        float v = 0.f;
        if (e < 16 * CIN) {
          int rm = e / CIN;
          int c  = e - rm * CIN;
          int row = rowbase + rm;
          int id = (row < n) ? pk[rm] : -1;
          if (id >= 0) v = fin[(size_t)id * CIN + c];
        }
        rg1[i] = v;
      }
    }
  };
  auto stStage = [&](float* buf) {
    if constexpr (VEC2) {
#pragma unroll
      for (int i = 0; i < NS; ++i) {
        int e2 = tid + i * 128;
        int rm = e2 / (CIN / 2);
        int c2 = e2 - rm * (CIN / 2);
        *(v2f*)(buf + rm * CINP + c2 * 2) = rg2[i];
      }
    } else {
#pragma unroll
      for (int i = 0; i < NS; ++i) {
        int e = tid + i * 128;
        if (e < 16 * CIN) {
          int rm = e / CIN;
          int c  = e - rm * CIN;
          buf[rm * CINP + c] = rg1[i];
        }
      }
    }
  };

  // ---- WMMA burst for one kernel offset -----------------------------------
  auto compute = [&](int k, const float* buf) {
    const float* __restrict__ Am = buf + m * CINP + 2 * hlf;
    const float* __restrict__ Wb =
        W + (size_t)k * CIN * COUT + (size_t)(2 * hlf) * COUT + cb + m;
    constexpr int NCH = CIN / 4;
#pragma unroll
    for (int ch = 0; ch < NCH; ++ch) {
      v2f a = *(const v2f*)(Am + ch * 4);            // A[m][k0..k0+1], b64 LDS
      v2f b0, b1;
      b0.x = Wb[(ch * 4 + 0) * COUT];                // B[k0][cb+m]
      b0.y = Wb[(ch * 4 + 1) * COUT];
      b1.x = Wb[(ch * 4 + 0) * COUT + 16];
      b1.y = Wb[(ch * 4 + 1) * COUT + 16];
      acc0 = __builtin_amdgcn_wmma_f32_16x16x4_f32(
          false, a, false, b0, (short)0, acc0, false, false);
      acc1 = __builtin_amdgcn_wmma_f32_16x16x4_f32(
          false, a, false, b1, (short)0, acc1, false, false);
    }
    if constexpr ((CIN % 4) != 0) {                  // only CIN==3
      int k0 = NCH * 4 + 2 * hlf;
      v2f a, b0, b1;
      a.x  = (k0     < CIN) ? Am[NCH * 4]     : 0.f;
      a.y  = (k0 + 1 < CIN) ? Am[NCH * 4 + 1] : 0.f;
      b0.x = (k0     < CIN) ? Wb[(NCH * 4 + 0) * COUT]      : 0.f;
      b0.y = (k0 + 1 < CIN) ? Wb[(NCH * 4 + 1) * COUT]      : 0.f;
      b1.x = (k0     < CIN) ? Wb[(NCH * 4 + 0) * COUT + 16] : 0.f;
      b1.y = (k0 + 1 < CIN) ? Wb[(NCH * 4 + 1) * COUT + 16] : 0.f;
      acc0 = __builtin_amdgcn_wmma_f32_16x16x4_f32(
          false, a, false, b0, (short)0, acc0, false, false);
      acc1 = __builtin_amdgcn_wmma_f32_16x16x4_f32(
          false, a, false, b1, (short)0, acc1, false, false);
    }
  };

  // ---- pipelined loop over the 27 kernel offsets --------------------------
  ldStage(0);
  stStage(bufA);
  __syncthreads();
#pragma unroll 1
  for (int k = 0; k < 27; ++k) {
    float* curb = (k & 1) ? bufB : bufA;
    float* nxtb = (k & 1) ? bufA : bufB;
    if (k < 26) ldStage(k + 1);                  // gathers in flight...
    if (active) {
      if (k < 26)                                // warm L2 for next W tile
        __builtin_prefetch(W + (size_t)(k + 1) * CIN * COUT + cb + m, 0, 3);
      compute(k, curb);                          // ...during the WMMA burst
    }
    if (k < 26) stStage(nxtb);
    __syncthreads();
  }

  if (active) {
#pragma unroll
    for (int j = 0; j < 8; ++j) {                // D: VGPR j -> row j (+8 hi half)
      int r = rowbase + j + (hlf << 3);
      if (r < n) {
        fout[(size_t)r * COUT + cb + m]      = acc0[j];
        fout[(size_t)r * COUT + cb + 16 + m] = acc1[j];
      }
    }
  }
}

static inline void conv_launch(int Ci, int Co, const float* fin, const float* W,
                               float* fout, const int* pairs, const int* dn,
                               int ub, hipStream_t s) {
  dim3 g((unsigned)((ub + 15) / 16), (unsigned)((Co + 127) / 128));
  int cinp = (Ci % 2) ? (Ci + 5) : (Ci + 4);
  size_t shmem = (size_t)2 * 16 * cinp * sizeof(float);
#define CONV_CASE(a, b)                                                        \
  if (Ci == a && Co == b) {                                                    \
    k_conv_wmma<a, b><<<g, 128, shmem, s>>>(fin, W, fout, pairs, dn, ub);      \
    return;                                                                    \
  }
  CONV_CASE(3, 64)    CONV_CASE(64, 64)   CONV_CASE(64, 96)   CONV_CASE(96, 96)
  CONV_CASE(96, 128)  CONV_CASE(128, 128) CONV_CASE(128, 160) CONV_CASE(160, 160)
  CONV_CASE(160, 192) CONV_CASE(192, 192) CONV_CASE(192, 224) CONV_CASE(224, 224)
  CONV_CASE(224, 256) CONV_CASE(256, 256)
#undef CONV_CASE
}

// ---------------- pooling: sorted-unique via dense prefix scan -------------

__device__ __forceinline__ unsigned encF(float f) {   // order-preserving uint
  unsigned o = __float_as_uint(f);
  return (o & 0x80000000u) ? ~o : (o | 0x80000000u);
}
__device__ __forceinline__ float decF(unsigned e) {
  unsigned o = (e & 0x80000000u) ? (e & 0x7fffffffu) : ~e;
  return __uint_as_float(o);
}

__global__ void k_pool_keys(const int* __restrict__ coords, int* __restrict__ keys,
                            int* __restrict__ occ, const int* __restrict__ pn, int S2) {
  int i = blockIdx.x * 256 + threadIdx.x;
  if (i >= *pn) return;
  int z = coords[i * 4 + 1] >> 1, y = coords[i * 4 + 2] >> 1, x = coords[i * 4 + 3] >> 1;
  int key = (z * S2 + y) * S2 + x;
  keys[i] = key;
  occ[key] = 1;
}

__global__ void k_scan1(const int* __restrict__ occ, int* __restrict__ rank,
                        int* __restrict__ bsum, int S3) {
  __shared__ int sh[256];
  int t = threadIdx.x;
  int base = blockIdx.x * 1024 + t * 4;
  int v0 = (base + 0 < S3) ? occ[base + 0] : 0;
  int v1 = (base + 1 < S3) ? occ[base + 1] : 0;
  int v2 = (base + 2 < S3) ? occ[base + 2] : 0;
  int v3 = (base + 3 < S3) ? occ[base + 3] : 0;
  int l1 = v0 + v1, l2 = l1 + v2, tot = l2 + v3;
  sh[t] = tot;
  __syncthreads();
  for (int off = 1; off < 256; off <<= 1) {
    int x = sh[t];
    int y = (t >= off) ? sh[t - off] : 0;
    __syncthreads();
    sh[t] = x + y;
    __syncthreads();
  }
  int excl = sh[t] - tot;
  if (base + 0 < S3) rank[base + 0] = excl;
  if (base + 1 < S3) rank[base + 1] = excl + v0;
  if (base + 2 < S3) rank[base + 2] = excl + l1;
  if (base + 3 < S3) rank[base + 3] = excl + l2;
  if (t == 255) bsum[blockIdx.x] = sh[255];
}

__global__ void k_scan2(const int* __restrict__ bsum, int* __restrict__ boff,
                        int nb, int* __restrict__ n_out) {
  __shared__ int sh[256];
  int t = threadIdx.x;
  int v = (t < nb) ? bsum[t] : 0;
  sh[t] = v;
  __syncthreads();
  for (int off = 1; off < 256; off <<= 1) {
    int x = sh[t];
    int y = (t >= off) ? sh[t - off] : 0;
    __syncthreads();
    sh[t] = x + y;
    __syncthreads();
  }
  boff[t] = sh[t] - v;
  if (t == 255) *n_out = sh[255];
}

__global__ void k_pool_coords(const int* __restrict__ occ, const int* __restrict__ rank,
                              const int* __restrict__ boff, int* __restrict__ ncoords,
                              int S3, int S2) {
  int s = blockIdx.x * 256 + threadIdx.x;
  if (s >= S3 || !occ[s]) return;
  int r = rank[s] + boff[s >> 10];
  int z = s / (S2 * S2);
  int rem = s - z * S2 * S2;
  ncoords[r * 4 + 0] = 0;
  ncoords[r * 4 + 1] = z;
  ncoords[r * 4 + 2] = rem / S2;
  ncoords[r * 4 + 3] = rem - (rem / S2) * S2;
}

__global__ void k_pool_scatter(const float* __restrict__ f, const int* __restrict__ keys,
                               const int* __restrict__ rank, const int* __restrict__ boff,
                               unsigned* __restrict__ enc, const int* __restrict__ pn, int C) {
  int n = *pn;
  int i = blockIdx.x * 256 + threadIdx.x;
  if (i >= n * C) return;
  int row = i / C;
  int c   = i - row * C;
  int key = keys[row];
  int r = rank[key] + boff[key >> 10];
  atomicMax(&enc[r * C + c], encF(f[i]));
}

__global__ void k_pool_decode(unsigned* __restrict__ enc, const int* __restrict__ pn, int C) {
  int n = *pn;
  int i = blockIdx.x * 256 + threadIdx.x;
  if (i >= n * C) return;
  float v = decF(enc[i]);
  ((float*)enc)[i] = v;
}

// ---------------------------------------------------------------------------

extern "C" void kernel_launch(void* const* d_in, const int* in_sizes, int n_in,
                              void* d_out, int out_size, void* d_ws, size_t ws_size,
                              hipStream_t stream) {
  const float* feats0 = (const float*)d_in[0];
  const int*   coors  = (const int*)d_in[1];
  const float* Wp[14];
  for (int i = 0; i < 14; ++i) Wp[i] = (const float*)d_in[3 + i];
  int n0 = in_sizes[0] / 3;

  static const int Cmid[7] = {64, 96, 128, 160, 192, 224, 256};
  static const int Cin0[7] = {3, 64, 96, 128, 160, 192, 224};
  int ub[7];
  ub[0] = n0;
  for (int l = 1; l < 7; ++l) {
    int S = 128 >> l;
    long s3 = (long)S * S * S;
    ub[l] = (ub[l - 1] < s3) ? ub[l - 1] : (int)s3;
  }

  char* ws = (char*)d_ws;
  auto carve = [&](size_t bytes) -> void* {
    void* p = (void*)ws;
    ws += (bytes + 255) & ~(size_t)255;
    return p;
  };
  int* dn    = (int*)carve(8 * sizeof(int));
  int* bsum  = (int*)carve(256 * sizeof(int));
  int* boff  = (int*)carve(256 * sizeof(int));
  int* grid  = (int*)carve((size_t)128 * 128 * 128 * sizeof(int));
  int* occ   = (int*)carve((size_t)64 * 64 * 64 * sizeof(int));
  int* rank  = (int*)carve((size_t)64 * 64 * 64 * sizeof(int));
  int* keys  = (int*)carve((size_t)n0 * sizeof(int));
  int* pairs = (int*)carve((size_t)27 * n0 * sizeof(int));
  int* cA    = (int*)carve((size_t)n0 * 4 * sizeof(int));
  int* cB    = (int*)carve((size_t)n0 * 4 * sizeof(int));
  size_t fbytes = 0;
  for (int l = 0; l < 7; ++l) {
    size_t b = (size_t)ub[l] * Cmid[l] * sizeof(float);
    if (b > fbytes) fbytes = b;
  }
  float* fbuf0 = (float*)carve(fbytes);
  float* fbuf1 = (float*)carve(fbytes);

  k_init_n<<<1, 32, 0, stream>>>(dn, n0);

  const int*   curC = coors;
  const float* curF = feats0;
  float* t0 = fbuf0;
  float* t1 = fbuf1;
  int* cuse = cA;
  int wi = 0;

  for (int l = 0; l < 7; ++l) {
    int S = 128 >> l;
    int S3 = S * S * S;
    int Ci = Cin0[l], Cm = Cmid[l];
    int gb = (ub[l] + 255) / 256;

    // neighbor structure for this level
    hipMemsetAsync(grid, 0xFF, (size_t)S3 * sizeof(int), stream);
    k_scatter_grid<<<gb, 256, 0, stream>>>(curC, grid, dn + l, S);
    k_build_pairs<<<dim3(gb, 27), 256, 0, stream>>>(curC, grid, pairs, dn + l, ub[l], S);

    // two SubMConv3d layers sharing the same pairs
    conv_launch(Ci, Cm, curF, Wp[wi], t0, pairs, dn + l, ub[l], stream);
    ++wi;
    float* dst2 = (l == 6) ? (float*)d_out : t1;
    conv_launch(Cm, Cm, t0, Wp[wi], dst2, pairs, dn + l, ub[l], stream);
    ++wi;

    if (l < 6) {
      int S2 = S >> 1;
      int S3n = S2 * S2 * S2;
      hipMemsetAsync(occ, 0, (size_t)S3n * sizeof(int), stream);
      k_pool_keys<<<gb, 256, 0, stream>>>(curC, keys, occ, dn + l, S2);
      int nb = (S3n + 1023) / 1024;
      k_scan1<<<nb, 256, 0, stream>>>(occ, rank, bsum, S3n);
      k_scan2<<<1, 256, 0, stream>>>(bsum, boff, nb, dn + l + 1);
      k_pool_coords<<<(S3n + 255) / 256, 256, 0, stream>>>(occ, rank, boff, cuse, S3n, S2);

      hipMemsetAsync(t0, 0, (size_t)ub[l + 1] * Cm * sizeof(float), stream);
      int tot  = ub[l] * Cm;
      int totn = ub[l + 1] * Cm;
      k_pool_scatter<<<(tot + 255) / 256, 256, 0, stream>>>(t1, keys, rank, boff,
                                                            (unsigned*)t0, dn + l, Cm);
      k_pool_decode<<<(totn + 255) / 256, 256, 0, stream>>>((unsigned*)t0, dn + l + 1, Cm);

      curF = t0;
      curC = cuse;
      cuse = (cuse == cA) ? cB : cA;
      float* tmp = t0; t0 = t1; t1 = tmp;
    }
  }
}